// RichardsonLucySubModule_79044578115775
// MI455X (gfx1250) — compile-verified
//
#include <hip/hip_runtime.h>
#include <hip/hip_bf16.h>

// ---------------------------------------------------------------------------
// Richardson-Lucy deconvolution for MI455X (gfx1250, wave32).
//
// Bandwidth-bound (~23.3 TB/s roof): fft2/ifft2 of 1024^2 complex planes, two
// 1-D passes each, every pass reads rows coalesced and writes TRANSPOSED so
// the column FFT is again a row FFT.  All pointwise ops (ROI padding, ratio,
// xK / xconj(K), relu-accumulate) are FUSED into FFT load/store phases, so a
// full fft2 -> xK -> ifft2 round trip per slice is exactly 4 plane passes.
//
// 1024-pt FFT = 16x16x4 Cooley-Tukey; both radix-16 stages are complex 16x16
// matmuls against DFT-16 executed with V_WMMA_F32_16X16X4_F32 (matrix pipe),
// radix-4 tail in VALU.  One wave owns one row (8KB LDS), 8 waves per block.
// Complex-plane loads use GLOBAL_LOAD_ASYNC_TO_LDS_B128 (memory->LDS direct,
// ASYNCcnt) on the contiguous 64KB block a workgroup consumes.
// ---------------------------------------------------------------------------

#define NXY     1024
#define NXYEXT  64
#define ROI     256          // ROISize
#define LOB     256          // Nxy/2 - ROI
#define HIB     768
#define NZ      24
#define ITN     2
#define INNER   896          // Nxy - 2*NxyExt
#define PI2     6.28318530717958647692f

typedef __attribute__((ext_vector_type(2))) float v2f;
typedef __attribute__((ext_vector_type(8))) float v8f;

static __device__ __forceinline__ v8f wmma4(v2f a, v2f b, v8f c) {
  // D(16x16,f32) = A(16x4,f32) * B(4x16,f32) + C  -- v_wmma_f32_16x16x4_f32
  // f32 NEG bits are C-neg only, so complex-matmul signs are handled by
  // negating the B fragment in registers.
  return __builtin_amdgcn_wmma_f32_16x16x4_f32(false, a, false, b, (short)0, c,
                                               false, false);
}

// Load modes: 0 = complex plane (async B128 -> LDS), 1 = obj slice padded to
// ROI (real), 2 = ratio plane from ImgExp/ImgEst (real).
// Store modes (always transposed: element k of row R -> out[k*NXY + R]):
// 0 = plain*scale, 1 = *K or *conj(K) (csign), 2 = ImgEst += relu(re/Ksum).
template <int LM, int SM>
__global__ __launch_bounds__(256) void fft1024(
    const void* __restrict__ inA,     // LM0: float2*, LM1: obj slice, LM2: ImgExp
    const float* __restrict__ inB,    // LM2: ImgEst
    const float* __restrict__ stats,  // [0]=Ksum (SM2), [1]=Tmp (LM2)
    void* __restrict__ vout,          // SM0/1: float2*, SM2: float* (ImgEst)
    const float* __restrict__ kr, const float* __restrict__ ki, float csign,
    float sdir, float scale) {
  __shared__ float2 lc[8][NXY];       // 64KB: 8 rows, interleaved complex
  const int tid  = threadIdx.x;
  const int row0 = blockIdx.x * 8;

  // ---------------- load phase ----------------
  if (LM == 0) {
    // contiguous 64KB block: rows row0..row0+7 of the complex plane
    unsigned lbase = (unsigned)(size_t)(void*)&lc[0][0];  // flat low32 = LDS off
    unsigned long long gbase =
        (unsigned long long)((const char*)inA + (size_t)row0 * NXY * 8);
#pragma unroll
    for (int t2 = 0; t2 < 16; ++t2) {
      unsigned off = (unsigned)((t2 * 256 + tid) * 16);
      asm volatile("global_load_async_to_lds_b128 %0, %1, %2"
                   :: "v"(lbase + off), "v"(off), "s"(gbase)
                   : "memory");
    }
    asm volatile("s_wait_asynccnt 0" ::: "memory");
  } else if (LM == 1) {
    const float* objz = (const float*)inA;
    for (int i = tid; i < 8 * NXY; i += 256) {
      int w = i >> 10, n = i & (NXY - 1);
      int y = row0 + w;
      float v = 0.0f;
      if (y >= LOB && y < HIB && n >= LOB && n < HIB)
        v = objz[(y - LOB) * (2 * ROI) + (n - LOB)];
      float2 o; o.x = v; o.y = 0.0f;
      lc[w][n] = o;
    }
  } else {  // LM == 2: ratio plane, border = 1
    const float* imgexp = (const float*)inA;
    const float eps = stats[1] * (1.0f / 200.0f);  // Tmp / SNR
    for (int i = tid; i < 8 * NXY; i += 256) {
      int w = i >> 10, n = i & (NXY - 1);
      int y = row0 + w;
      float v = 1.0f;
      if (y >= NXYEXT && y < NXY - NXYEXT && n >= NXYEXT && n < NXY - NXYEXT) {
        int idx = y * NXY + n;
        v = imgexp[idx] / (inB[idx] + eps);
      }
      float2 o; o.x = v; o.y = 0.0f;
      lc[w][n] = o;
    }
  }
  __syncthreads();

  // ---------------- per-wave 1024-pt FFT on LDS row ----------------
  const int w    = tid >> 5;          // wave id = row within block (wave32)
  const int lane = tid & 31;
  const int m    = lane & 15;         // M (or N) coordinate in fragment
  const int koff = (lane >> 4) << 1;  // lanes 16-31 hold K+2
  float2* L = lc[w];

  // A fragments of DFT-16: W[m][k] = e^{sdir*i*2pi*m*k/16}, K chunked by 4.
  float arv[4][2], aiv[4][2];
#pragma unroll
  for (int kc = 0; kc < 4; ++kc) {
#pragma unroll
    for (int e = 0; e < 2; ++e) {
      int k = 4 * kc + koff + e;
      float s, c;
      __sincosf(PI2 * (float)(m * k) * (1.0f / 16.0f), &s, &c);
      arv[kc][e] = c;
      aiv[kc][e] = sdir * s;
    }
  }

  // Stage A: y[k1][n2] = DFT16 over n1 of x[n2 + 64 n1]; twiddle w1024^{k1 n2}.
  for (int b = 0; b < 4; ++b) {
    const int col = 16 * b + m;
    v8f cr = {0, 0, 0, 0, 0, 0, 0, 0};
    v8f ci = {0, 0, 0, 0, 0, 0, 0, 0};
#pragma unroll
    for (int kc = 0; kc < 4; ++kc) {
      int t0 = 4 * kc + koff;
      float2 x0 = L[col + (t0 << 6)];
      float2 x1 = L[col + ((t0 + 1) << 6)];
      v2f ar = {arv[kc][0], arv[kc][1]};
      v2f ai = {aiv[kc][0], aiv[kc][1]};
      v2f br = {x0.x, x1.x};
      v2f bi = {x0.y, x1.y};
      cr = wmma4(ar, br, cr);
      cr = wmma4(ai, -bi, cr);   // Re: Wr*Xr - Wi*Xi
      ci = wmma4(ar, bi, ci);
      ci = wmma4(ai, br, ci);    // Im: Wr*Xi + Wi*Xr
    }
#pragma unroll
    for (int g = 0; g < 8; ++g) {            // D rows: M = g + 8*(lane>=16)
      int k1 = g + ((lane >> 4) << 3);
      float s, c;
      __sincosf(PI2 * (float)(k1 * col) * (1.0f / 1024.0f), &s, &c);
      float tr = c, ti = sdir * s;
      float2 o;
      o.x = cr[g] * tr - ci[g] * ti;
      o.y = cr[g] * ti + ci[g] * tr;
      L[col + (k1 << 6)] = o;
    }
  }

  // Stage B: DFT16 over s of y[k1][4s+q]; twiddle w64^{p q}; z at q+4p+64k1.
  for (int b = 0; b < 4; ++b) {
    const int cg = 16 * b + m;
    const int k1 = cg >> 2;
    const int q  = cg & 3;
    v8f cr = {0, 0, 0, 0, 0, 0, 0, 0};
    v8f ci = {0, 0, 0, 0, 0, 0, 0, 0};
#pragma unroll
    for (int kc = 0; kc < 4; ++kc) {
      int s0 = 4 * kc + koff;
      float2 x0 = L[((s0 << 2) + q) + (k1 << 6)];
      float2 x1 = L[(((s0 + 1) << 2) + q) + (k1 << 6)];
      v2f ar = {arv[kc][0], arv[kc][1]};
      v2f ai = {aiv[kc][0], aiv[kc][1]};
      v2f br = {x0.x, x1.x};
      v2f bi = {x0.y, x1.y};
      cr = wmma4(ar, br, cr);
      cr = wmma4(ai, -bi, cr);
      ci = wmma4(ar, bi, ci);
      ci = wmma4(ai, br, ci);
    }
#pragma unroll
    for (int g = 0; g < 8; ++g) {
      int p = g + ((lane >> 4) << 3);
      float s, c;
      __sincosf(PI2 * (float)(p * q) * (1.0f / 64.0f), &s, &c);
      float tr = c, ti = sdir * s;
      float2 o;
      o.x = cr[g] * tr - ci[g] * ti;
      o.y = cr[g] * ti + ci[g] * tr;
      L[q + (p << 2) + (k1 << 6)] = o;
    }
  }

  // Stage C: radix-4 over q; out index k = k1 + 16p + 256r.
  // Load everything first (DS ops in-order per wave), then store: outputs
  // alias the z addresses across lanes.
  float2 z[8][4];
#pragma unroll
  for (int u = 0; u < 8; ++u) {
    int idx = lane + (u << 5);
    int k1 = idx & 15, p = idx >> 4;
#pragma unroll
    for (int q = 0; q < 4; ++q) z[u][q] = L[q + (p << 2) + (k1 << 6)];
  }
#pragma unroll
  for (int u = 0; u < 8; ++u) {
    int idx = lane + (u << 5);
    int k1 = idx & 15, p = idx >> 4;
    float t0r = z[u][0].x + z[u][2].x, t0i = z[u][0].y + z[u][2].y;
    float t1r = z[u][0].x - z[u][2].x, t1i = z[u][0].y - z[u][2].y;
    float t2r = z[u][1].x + z[u][3].x, t2i = z[u][1].y + z[u][3].y;
    float t3r = z[u][1].x - z[u][3].x, t3i = z[u][1].y - z[u][3].y;
    int base = k1 + (p << 4);
    float2 o0, o1, o2, o3;
    o0.x = t0r + t2r;            o0.y = t0i + t2i;             // r=0
    o1.x = t1r - sdir * t3i;     o1.y = t1i + sdir * t3r;      // r=1
    o2.x = t0r - t2r;            o2.y = t0i - t2i;             // r=2
    o3.x = t1r + sdir * t3i;     o3.y = t1i - sdir * t3r;      // r=3
    L[base]       = o0;
    L[base + 256] = o1;
    L[base + 512] = o2;
    L[base + 768] = o3;
  }
  __syncthreads();

  // ---------------- transposed store phase (fused pointwise) ----------------
  if (SM == 2) {
    float* fo = (float*)vout;                      // ImgEst accumulate
    const float rk = 1.0f / stats[0];
    for (int i = tid; i < 8 * NXY; i += 256) {
      int ww = i & 7, k = i >> 3;
      int idx = k * NXY + row0 + ww;
      fo[idx] += fmaxf(lc[ww][k].x * scale * rk, 0.0f);
    }
  } else if (SM == 1) {
    float2* fo = (float2*)vout;                    // * K / conj(K)
    for (int i = tid; i < 8 * NXY; i += 256) {
      int ww = i & 7, k = i >> 3;
      int idx = k * NXY + row0 + ww;               // == (k2,k1) of fft2 == K idx
      float2 v = lc[ww][k];
      float c = kr[idx], d = csign * ki[idx];
      float2 o;
      o.x = (v.x * c - v.y * d) * scale;
      o.y = (v.x * d + v.y * c) * scale;
      fo[idx] = o;
    }
  } else {
    float2* fo = (float2*)vout;                    // plain * scale
    for (int i = tid; i < 8 * NXY; i += 256) {
      int ww = i & 7, k = i >> 3;
      float2 v = lc[ww][k];
      float2 o; o.x = v.x * scale; o.y = v.y * scale;
      fo[k * NXY + row0 + ww] = o;
    }
  }
}

// ------------------------- reduction / small kernels ------------------------

// stats layout (float*/unsigned*):
// [0]=Ksum  [1]=Tmp(median)  [2]=min bits  [3]=max bits  [4..4+1024)=hist
__global__ void stats_zero_k(float* stats) {
  unsigned* su = (unsigned*)stats;
  for (int i = threadIdx.x; i < 1028; i += blockDim.x) su[i] = 0u;
  if (threadIdx.x == 0) {
    stats[0] = 0.0f;
    stats[1] = 0.0f;
    su[2] = 0x7F7FFFFFu;  // +FLT_MAX bits (values >= 0)
    su[3] = 0u;
  }
}

__global__ void hist_reset_k(float* stats) {
  unsigned* su = (unsigned*)stats;
  int i = blockIdx.x * blockDim.x + threadIdx.x;
  if (i < 1024) su[4 + i] = 0u;
  if (i == 0) { su[2] = 0x7F7FFFFFu; su[3] = 0u; }
}

__global__ void zero_f32_k(float* p, int n) {
  for (int i = blockIdx.x * blockDim.x + threadIdx.x; i < n;
       i += gridDim.x * blockDim.x)
    p[i] = 0.0f;
}

__global__ void ksum_reduce_k(const float* __restrict__ x, int n, float* stats) {
  __shared__ float sm[256];
  float acc = 0.0f;
  for (int i = blockIdx.x * blockDim.x + threadIdx.x; i < n;
       i += gridDim.x * blockDim.x)
    acc += x[i];
  sm[threadIdx.x] = acc;
  __syncthreads();
  for (int s = 128; s > 0; s >>= 1) {
    if (threadIdx.x < s) sm[threadIdx.x] += sm[threadIdx.x + s];
    __syncthreads();
  }
  if (threadIdx.x == 0) atomicAdd(&stats[0], sm[0]);
}

// ImgExp = pad(relu(imstack - 110), 64 each side)
__global__ void imgexp_pad_k(const float* __restrict__ img, float* __restrict__ out) {
  for (int i = blockIdx.x * blockDim.x + threadIdx.x; i < NXY * NXY;
       i += gridDim.x * blockDim.x) {
    int y = i >> 10, x = i & (NXY - 1);
    float v = 0.0f;
    if (y >= NXYEXT && y < NXY - NXYEXT && x >= NXYEXT && x < NXY - NXYEXT)
      v = fmaxf(img[(y - NXYEXT) * INNER + (x - NXYEXT)] - 110.0f, 0.0f);
    out[i] = v;
  }
}

__global__ void minmax_k(const float* __restrict__ x, float* stats) {
  unsigned* su = (unsigned*)stats;
  unsigned lo = 0x7F7FFFFFu, hi = 0u;
  for (int i = blockIdx.x * blockDim.x + threadIdx.x; i < NXY * NXY;
       i += gridDim.x * blockDim.x) {
    unsigned b = __float_as_uint(x[i]);   // x >= 0: uint order == float order
    lo = (b < lo) ? b : lo;
    hi = (b > hi) ? b : hi;
  }
  atomicMin(&su[2], lo);
  atomicMax(&su[3], hi);
}

__global__ void hist_k(const float* __restrict__ x, float* stats) {
  __shared__ unsigned hh[1024];
  unsigned* su = (unsigned*)stats;
  for (int i = threadIdx.x; i < 1024; i += blockDim.x) hh[i] = 0u;
  __syncthreads();
  const float vmin = __uint_as_float(su[2]);
  const float vmax = __uint_as_float(su[3]);
  const float wdt = (vmax - vmin) * (1.0f / 1024.0f) + 1e-30f;
  const float rw = 1.0f / wdt;
  for (int i = blockIdx.x * blockDim.x + threadIdx.x; i < NXY * NXY;
       i += gridDim.x * blockDim.x) {
    int b = (int)((x[i] - vmin) * rw);
    b = (b < 0) ? 0 : ((b > 1023) ? 1023 : b);
    atomicAdd(&hh[b], 1u);
  }
  __syncthreads();
  for (int i = threadIdx.x; i < 1024; i += blockDim.x)
    if (hh[i]) atomicAdd(&su[4 + i], hh[i]);
}

// approximate median = center of the bin holding the middle element
__global__ void median_k(float* stats) {
  if (threadIdx.x != 0) return;
  unsigned* su = (unsigned*)stats;
  const float vmin = __uint_as_float(su[2]);
  const float vmax = __uint_as_float(su[3]);
  const float wdt = (vmax - vmin) * (1.0f / 1024.0f) + 1e-30f;
  unsigned long long half = (unsigned long long)(NXY * NXY) / 2ull;
  unsigned long long acc = 0;
  float tmp = vmin;
  for (int b = 0; b < 1024; ++b) {
    acc += su[4 + b];
    if (acc > half) { tmp = vmin + ((float)b + 0.5f) * wdt; break; }
  }
  stats[1] = tmp;
}

// obj *= relu(real(back)/Ksum) on the ROI, broadcast over Nz
__global__ void obj_update_k(const float2* __restrict__ back,
                             const float* __restrict__ stats,
                             float* __restrict__ obj) {
  const float rk = 1.0f / stats[0];
  const int n = NZ * 2 * ROI * 2 * ROI;
  for (int j = blockIdx.x * blockDim.x + threadIdx.x; j < n;
       j += gridDim.x * blockDim.x) {
    int rem = j & (2 * ROI * 2 * ROI - 1);
    int y = rem >> 9, x = rem & (2 * ROI - 1);
    float b = fmaxf(back[(y + LOB) * NXY + (x + LOB)].x * rk, 0.0f);
    obj[j] *= b;
  }
}

// ---------------------------------------------------------------------------

extern "C" void kernel_launch(void* const* d_in, const int* in_sizes, int n_in,
                              void* d_out, int out_size, void* d_ws, size_t ws_size,
                              hipStream_t stream) {
  (void)in_sizes; (void)n_in; (void)out_size; (void)ws_size;

  const float* imstack = (const float*)d_in[0];   // [1,1,896,896]
  const float* initv   = (const float*)d_in[1];   // [1,24,512,512]
  const float* convr   = (const float*)d_in[2];   // [1,1,1024,1024]
  const float* convi   = (const float*)d_in[3];   // [1,1,1024,1024]
  float* obj = (float*)d_out;                     // [1,24,512,512], live object

  // workspace carve-out (~24 MB)
  float* ws = (float*)d_ws;
  float2* bufA  = (float2*)ws;                    // 8 MB
  float2* bufB  = bufA + NXY * NXY;               // 8 MB
  float* imgest = (float*)(bufB + NXY * NXY);     // 4 MB
  float* imgexp = imgest + NXY * NXY;             // 4 MB
  float* stats  = imgexp + NXY * NXY;             // 1028 words

  const int PB = 256;
  const int PG = 2048;
  const float inv = 1.0f / 1024.0f;

  hipMemcpyAsync(obj, initv, sizeof(float) * NZ * 2 * ROI * 2 * ROI,
                 hipMemcpyDeviceToDevice, stream);
  stats_zero_k<<<1, 256, 0, stream>>>(stats);
  ksum_reduce_k<<<PG, PB, 0, stream>>>(convr, NXY * NXY, stats);
  imgexp_pad_k<<<PG, PB, 0, stream>>>(imstack, imgexp);

  for (int it = 0; it < ITN; ++it) {
    zero_f32_k<<<PG, PB, 0, stream>>>(imgest, NXY * NXY);

    // ---- forward projection: fft2 -> *K -> ifft2, fully fused, 4 passes ----
    for (int z = 0; z < NZ; ++z) {
      const float* objz = obj + z * (2 * ROI) * (2 * ROI);
      fft1024<1, 0><<<128, 256, 0, stream>>>(objz, nullptr, stats, bufB,
                                             nullptr, nullptr, 0.f, -1.f, 1.f);
      fft1024<0, 1><<<128, 256, 0, stream>>>(bufB, nullptr, stats, bufA,
                                             convr, convi, 1.f, -1.f, 1.f);
      fft1024<0, 0><<<128, 256, 0, stream>>>(bufA, nullptr, stats, bufB,
                                             nullptr, nullptr, 0.f, 1.f, inv);
      fft1024<0, 2><<<128, 256, 0, stream>>>(bufB, nullptr, stats, imgest,
                                             nullptr, nullptr, 0.f, 1.f, inv);
    }

    // ---- median(ImgEst): min/max + 1024-bin histogram ----
    hist_reset_k<<<4, 256, 0, stream>>>(stats);
    minmax_k<<<PG, PB, 0, stream>>>(imgest, stats);
    hist_k<<<PG, PB, 0, stream>>>(imgest, stats);
    median_k<<<1, 32, 0, stream>>>(stats);

    // ---- back projection: ratio fused into P1, conj(K) fused into P2 ----
    fft1024<2, 0><<<128, 256, 0, stream>>>(imgexp, imgest, stats, bufB,
                                           nullptr, nullptr, 0.f, -1.f, 1.f);
    fft1024<0, 1><<<128, 256, 0, stream>>>(bufB, nullptr, stats, bufA,
                                           convr, convi, -1.f, -1.f, 1.f);
    fft1024<0, 0><<<128, 256, 0, stream>>>(bufA, nullptr, stats, bufB,
                                           nullptr, nullptr, 0.f, 1.f, inv);
    fft1024<0, 0><<<128, 256, 0, stream>>>(bufB, nullptr, stats, bufA,
                                           nullptr, nullptr, 0.f, 1.f, inv);
    obj_update_k<<<PG, PB, 0, stream>>>(bufA, stats, obj);
  }
}